// AttentionStem_429496729731
// MI455X (gfx1250) — compile-verified
//
#include <hip/hip_runtime.h>
#include <hip/hip_bf16.h>

// AttentionStem on MI455X (gfx1250).
// Bandwidth-bound op (16MB out / ~0.8MB in => ~0.75us floor at 23.3TB/s).
// Strategy: fold the emb-softmax + value branches into 16 effective 1x1 convs
// Wv[k] (prep kernel), then per 16-pixel wave-tile run the three channel
// contractions (q, key, vsum) as V_WMMA_F32_16X16X4_F32 (M=16 pixels,
// K=IC padded to 4, N=16 channels) and an online softmax over the 16 window
// slots, entirely in the WMMA C-layout registers.

#define KS   4
#define PAD  2
#define MB   4
#define OCH  64
#define ICH  3
#define HH   128
#define WW   128
#define BB   4

typedef __attribute__((ext_vector_type(2))) float v2f;
typedef __attribute__((ext_vector_type(8))) float v8f;

// ---------------------------------------------------------------------------
// Prep: emb softmax (M x 16), folded value weights Wv[16][64][4] (c=3 zero),
// and K-padded key/query weights (64 x 4). All into ws:
//   ws[0..4095]    = Wv[k][o][c4]
//   ws[4096..4351] = keyp[o][c4]
//   ws[4352..4607] = qryp[o][c4]
// ---------------------------------------------------------------------------
__global__ void attnstem_prep(const float* __restrict__ key_w,
                              const float* __restrict__ query_w,
                              const float* __restrict__ value_w,
                              const float* __restrict__ emb_a,
                              const float* __restrict__ emb_b,
                              const float* __restrict__ emb_mix,
                              float* __restrict__ ws) {
  __shared__ float s_emb[MB][16];
  const int t = threadIdx.x;
  if (t < 16) {
    const int k = t, i = k >> 2, j = k & 3;
    float l[MB];
    float mx = -1e30f;
    for (int m = 0; m < MB; ++m) {
      float la = 0.f, lb = 0.f;
      for (int o = 0; o < OCH; ++o) {
        la += emb_mix[m * OCH + o] * emb_a[o * KS + i];
        lb += emb_mix[m * OCH + o] * emb_b[o * KS + j];
      }
      l[m] = la + lb;
      mx = fmaxf(mx, l[m]);
    }
    float ssum = 0.f;
    for (int m = 0; m < MB; ++m) { l[m] = __expf(l[m] - mx); ssum += l[m]; }
    for (int m = 0; m < MB; ++m) s_emb[m][k] = l[m] / ssum;
  }
  __syncthreads();
  for (int idx = t; idx < 16 * OCH * 4; idx += blockDim.x) {
    const int c = idx & 3;
    const int o = (idx >> 2) & (OCH - 1);
    const int k = idx >> 8;
    float v = 0.f;
    if (c < ICH) {
      for (int m = 0; m < MB; ++m)
        v += s_emb[m][k] * value_w[(m * OCH + o) * ICH + c];
    }
    ws[idx] = v;
  }
  for (int idx = t; idx < OCH * 4; idx += blockDim.x) {
    const int c = idx & 3;
    const int o = idx >> 2;
    ws[4096 + idx] = (c < ICH) ? key_w[o * ICH + c] : 0.f;
    ws[4352 + idx] = (c < ICH) ? query_w[o * ICH + c] : 0.f;
  }
}

// ---------------------------------------------------------------------------
// Main kernel: 8 waves/block, each wave owns 16 pixels of one image row and
// computes all 64 output channels for them.
// ---------------------------------------------------------------------------
__global__ __launch_bounds__(256) void attnstem_main(
    const float* __restrict__ x, const float* __restrict__ ws,
    float* __restrict__ out) {
  // per-wave x tile: [c:4][r:4][col:20] floats (c=3 and col=19 are padding)
  __shared__ float tile[8][4 * 4 * 20];

  const int lane = threadIdx.x & 31;
  const int wave = threadIdx.x >> 5;
  const int task = blockIdx.x * 8 + wave;      // 4096 tasks total
  const int b    = task >> 10;                 // 1024 tasks per image
  const int rem  = task & 1023;
  const int h    = rem >> 3;                   // output row
  const int wbase = (rem & 7) << 4;            // first output col of tile

  float* T = tile[wave];
  // Stage padded x tile (padded rows h..h+3, padded cols wbase..wbase+18).
  for (int t = lane; t < 320; t += 32) {
    const int c   = t / 80;
    const int r   = (t - c * 80) / 20;
    const int col = t % 20;
    const int y   = h + r - PAD;
    const int xx  = wbase + col - PAD;
    float v = 0.f;
    if (c < ICH && (unsigned)y < (unsigned)HH && (unsigned)xx < (unsigned)WW)
      v = x[((b * ICH + c) * HH + y) * WW + xx];
    T[t] = v;
  }
  __syncthreads();

  const int m  = lane & 15;          // pixel index within tile (A rows)
  const int cb = (lane >> 4) << 1;   // K-pair: lanes 0-15 -> c{0,1}, 16-31 -> c{2,3}

  // A operand per window slot k=(i,j): 16 pixels x 4 channels (f32).
  v2f A[16];
#pragma unroll
  for (int k = 0; k < 16; ++k) {
    const int i = k >> 2, j = k & 3;
    A[k].x = T[cb * 80 + i * 20 + j + m];
    A[k].y = T[(cb + 1) * 80 + i * 20 + j + m];
  }

  const float* Wv   = ws;
  const float* keyp = ws + 4096;
  const float* qryp = ws + 4352;

#pragma unroll
  for (int g = 0; g < 4; ++g) {                // 16-channel N-groups
    const int n = g * 16 + m;                  // output channel (B cols)
    const v2f bq = *(const v2f*)(qryp + n * 4 + cb);
    const v2f bk = *(const v2f*)(keyp + n * 4 + cb);
    v8f zero = {};
    // q at the window center == slot (i=2, j=2) == k=10 of the padded window
    v8f q = __builtin_amdgcn_wmma_f32_16x16x4_f32(
        false, A[10], false, bq, (short)0, zero, false, false);

    v8f rm, Z, acc;
#pragma unroll
    for (int e = 0; e < 8; ++e) { rm[e] = -1e30f; Z[e] = 0.f; acc[e] = 0.f; }

#pragma unroll
    for (int k = 0; k < 16; ++k) {
      const v2f bv = *(const v2f*)(Wv + (k * OCH + n) * 4 + cb);
      v8f kv = __builtin_amdgcn_wmma_f32_16x16x4_f32(
          false, A[k], false, bk, (short)0, zero, false, false);
      v8f vs = __builtin_amdgcn_wmma_f32_16x16x4_f32(
          false, A[k], false, bv, (short)0, zero, false, false);
#pragma unroll
      for (int e = 0; e < 8; ++e) {          // online softmax, elementwise in C layout
        const float s  = q[e] * kv[e];
        const float nm = fmaxf(rm[e], s);
        const float sc = __expf(rm[e] - nm);
        const float es = __expf(s - nm);
        Z[e]   = Z[e] * sc + es;
        acc[e] = acc[e] * sc + es * vs[e];
        rm[e]  = nm;
      }
    }

    // D layout: lane -> channel n, VGPR e -> pixel m = e + 8*(lane>=16)
    float* op = out + (((b * OCH + n) * HH + h) * WW) + wbase + ((lane >> 4) << 3);
    float4 o0 = make_float4(acc[0] / Z[0], acc[1] / Z[1], acc[2] / Z[2], acc[3] / Z[3]);
    float4 o1 = make_float4(acc[4] / Z[4], acc[5] / Z[5], acc[6] / Z[6], acc[7] / Z[7]);
    *(float4*)(op)     = o0;
    *(float4*)(op + 4) = o1;
  }
}

extern "C" void kernel_launch(void* const* d_in, const int* in_sizes, int n_in,
                              void* d_out, int out_size, void* d_ws, size_t ws_size,
                              hipStream_t stream) {
  const float* x       = (const float*)d_in[0];
  const float* key_w   = (const float*)d_in[1];
  const float* query_w = (const float*)d_in[2];
  const float* value_w = (const float*)d_in[3];
  const float* emb_a   = (const float*)d_in[4];
  const float* emb_b   = (const float*)d_in[5];
  const float* emb_mix = (const float*)d_in[6];
  float* out = (float*)d_out;
  float* ws  = (float*)d_ws;

  attnstem_prep<<<1, 256, 0, stream>>>(key_w, query_w, value_w,
                                       emb_a, emb_b, emb_mix, ws);
  // 4096 wave-tasks (B=4 * H=128 * W/16=8), 8 waves per 256-thread block
  attnstem_main<<<512, 256, 0, stream>>>(x, ws, out);
}